// TripletModel_47132971106315
// MI455X (gfx1250) — compile-verified
//
#include <hip/hip_runtime.h>
#include <math.h>

#define NN 8192
#define DD 128
#define FMAXV 3.402823466e38f
#define MARGINV 1.0f
#define EPSV 1e-6f

#define NCHUNK 8                 // column chunks (grid.y)
#define CHUNKCOLS (NN / NCHUNK)  // 1024 columns per chunk
#define TILES (CHUNKCOLS / 16)   // 64 col-tiles per chunk
#define BPAD 132                 // padded dwords per column in LDS (bank-conflict-free)
#define BUFELEMS (16 * BPAD)     // one B tile in LDS (floats)

typedef float v2f __attribute__((ext_vector_type(2)));
typedef float v8f __attribute__((ext_vector_type(8)));

// ---------------------------------------------------------------------------
// Kernel A: per-row squared norms. One wave per row, lane handles 4 floats.
// ---------------------------------------------------------------------------
__global__ __launch_bounds__(256) void row_norms_kernel(
    const float* __restrict__ embed, float* __restrict__ normsq)
{
    const int wave = (blockIdx.x * blockDim.x + threadIdx.x) >> 5;
    const int lane = threadIdx.x & 31;
    if (wave >= NN) return;
    const float4 v = *(const float4*)(embed + (size_t)wave * DD + lane * 4);
    float s = v.x * v.x + v.y * v.y + v.z * v.z + v.w * v.w;
#pragma unroll
    for (int off = 16; off > 0; off >>= 1) s += __shfl_xor(s, off, 32);
    if (lane == 0) normsq[wave] = s;
}

// ---------------------------------------------------------------------------
// Kernel B: fused Gram-tile WMMA + batch-hard mining.
// Block = 128 anchor rows (wave w owns rows base+w*16) x one 1024-col chunk.
// All 8 waves share each 16-col B tile via double-buffered LDS.
//
// V_WMMA_F32_16X16X4_F32 layouts (ISA 7.12.2):
//   A (16x4):  lane L -> M = L%16; lanes 0-15 hold K={4t,4t+1}, lanes 16-31
//              hold K={4t+2,4t+3} in (v0,v1).
//   B (4x16):  same pattern with N = L%16 (Gram: B columns are embed rows).
//   C/D:       vgpr r -> M = r + 8*(lane>=16), N = lane%16.
// ---------------------------------------------------------------------------
__global__ __launch_bounds__(256) void mine_kernel(
    const float* __restrict__ embed,
    const int*   __restrict__ labels,
    const float* __restrict__ normsq,
    float* __restrict__ pposv, int* __restrict__ pposi,
    float* __restrict__ pnegv, int* __restrict__ pnegi)
{
    __shared__ float sh_b[2][BUFELEMS];

    const int tid   = threadIdx.x;
    const int lane  = tid & 31;
    const int w     = tid >> 5;
    const int half  = (lane >> 4) & 1;   // 0: lanes 0-15, 1: lanes 16-31
    const int l16   = lane & 15;
    const int chunk = blockIdx.y;
    const int chunkbase   = chunk * CHUNKCOLS;
    const int waveRowBase = blockIdx.x * 128 + w * 16;

    // Cooperative-stage index math: 512 float4 per tile, 2 per thread.
    const int f0 = tid, f1 = tid + 256;
    const int c0 = f0 >> 5, o0 = (f0 & 31) << 2;   // col-in-tile, dword offset
    const int c1 = f1 >> 5, o1 = (f1 & 31) << 2;

    // Loop-invariant A operand: 32 K-pairs for this lane's row slice.
    v2f a[32];
    {
        const float* ap = embed + (size_t)(waveRowBase + l16) * DD + half * 2;
#pragma unroll
        for (int t = 0; t < 32; ++t) a[t] = *(const v2f*)(ap + 4 * t);
    }

    // Loop-invariant per-row norms / labels for this lane's 8 M positions.
    float normA[8];
    int   labA[8];
#pragma unroll
    for (int r = 0; r < 8; ++r) {
        const int row = waveRowBase + r + half * 8;
        normA[r] = normsq[row];
        labA[r]  = labels[row];
    }

    // Running (value, index) mining state per M position.
    float pv[8], nv[8];
    int   pi[8], ni[8];
#pragma unroll
    for (int r = 0; r < 8; ++r) {
        pv[r] = -FMAXV; pi[r] = 0x7fffffff;
        nv[r] =  FMAXV; ni[r] = 0x7fffffff;
    }

    // Stage tile 0 into LDS buffer 0.
    {
        const float* g = embed + (size_t)chunkbase * DD;
        const float4 q0 = *(const float4*)(g + (size_t)c0 * DD + o0);
        const float4 q1 = *(const float4*)(g + (size_t)c1 * DD + o1);
        *(float4*)(&sh_b[0][c0 * BPAD + o0]) = q0;
        *(float4*)(&sh_b[0][c1 * BPAD + o1]) = q1;
    }
    __syncthreads();

    for (int it = 0; it < TILES; ++it) {
        const int colbase = chunkbase + it * 16;
        const int buf = it & 1;
        const bool hasNext = (it + 1 < TILES);

        // Prefetch next B tile into registers (latency hides behind compute).
        float4 q0, q1;
        if (hasNext) {
            const float* g = embed + (size_t)(colbase + 16) * DD;
            q0 = *(const float4*)(g + (size_t)c0 * DD + o0);
            q1 = *(const float4*)(g + (size_t)c1 * DD + o1);
        }

        // 16x16 Gram tile: 32 chained f32 WMMAs, two independent acc chains.
        const float* bs = &sh_b[buf][l16 * BPAD + half * 2];
        v8f acc0 = {0.f, 0.f, 0.f, 0.f, 0.f, 0.f, 0.f, 0.f};
        v8f acc1 = {0.f, 0.f, 0.f, 0.f, 0.f, 0.f, 0.f, 0.f};
#pragma unroll
        for (int s = 0; s < 16; ++s) {
            const v2f b0 = *(const v2f*)(bs + 8 * s);
            const v2f b1 = *(const v2f*)(bs + 8 * s + 4);
            acc0 = __builtin_amdgcn_wmma_f32_16x16x4_f32(
                false, a[2 * s], false, b0, (short)0, acc0, false, false);
            acc1 = __builtin_amdgcn_wmma_f32_16x16x4_f32(
                false, a[2 * s + 1], false, b1, (short)0, acc1, false, false);
        }

        // Mining update for this tile's 16 columns.
        const int   col = colbase + l16;
        const float nb  = normsq[col];
        const int   lb  = labels[col];
#pragma unroll
        for (int r = 0; r < 8; ++r) {
            const float dot  = acc0[r] + acc1[r];
            const float dist = normA[r] + nb - 2.0f * dot;
            const bool  sim  = (labA[r] == lb);
            // hardest positive: argmax(dist*sim), first index on ties
            const float p = sim ? dist : 0.0f;
            if (p > pv[r]) { pv[r] = p; pi[r] = col; }
            // hardest negative: argmin(neg + (neg<=0)*FMAX), first index on ties
            float nm = sim ? 0.0f : dist;
            if (nm <= 0.0f) nm += FMAXV;
            if (nm < nv[r]) { nv[r] = nm; ni[r] = col; }
        }

        // Rotate the double buffer.
        if (hasNext) {
            __syncthreads();
            *(float4*)(&sh_b[1 - buf][c0 * BPAD + o0]) = q0;
            *(float4*)(&sh_b[1 - buf][c1 * BPAD + o1]) = q1;
            __syncthreads();
        }
    }

    // Reduce across the 16 lanes sharing each row (width-16 xor shuffles).
#pragma unroll
    for (int r = 0; r < 8; ++r) {
#pragma unroll
        for (int m = 1; m < 16; m <<= 1) {
            float ov = __shfl_xor(pv[r], m, 16);
            int   oi = __shfl_xor(pi[r], m, 16);
            if (ov > pv[r] || (ov == pv[r] && oi < pi[r])) { pv[r] = ov; pi[r] = oi; }
            float on = __shfl_xor(nv[r], m, 16);
            int   oj = __shfl_xor(ni[r], m, 16);
            if (on < nv[r] || (on == nv[r] && oj < ni[r])) { nv[r] = on; ni[r] = oj; }
        }
    }

    // Each wave owns its 16 rows outright: write per-chunk partials.
    if (l16 == 0) {
#pragma unroll
        for (int r = 0; r < 8; ++r) {
            const int row = waveRowBase + r + half * 8;
            const size_t idx = (size_t)chunk * NN + row;
            pposv[idx] = pv[r]; pposi[idx] = pi[r];
            pnegv[idx] = nv[r]; pnegi[idx] = ni[r];
        }
    }
}

// ---------------------------------------------------------------------------
// Kernel B2: merge chunk partials (ascending chunk order + idx tie-break
// reproduces global first-index argmax/argmin).
// ---------------------------------------------------------------------------
__global__ __launch_bounds__(256) void combine_kernel(
    const float* __restrict__ pposv, const int* __restrict__ pposi,
    const float* __restrict__ pnegv, const int* __restrict__ pnegi,
    int* __restrict__ hp, int* __restrict__ hn)
{
    const int row = blockIdx.x * blockDim.x + threadIdx.x;
    if (row >= NN) return;
    float bpv = -FMAXV; int bpi = 0x7fffffff;
    float bnv =  FMAXV; int bni = 0x7fffffff;
#pragma unroll
    for (int c = 0; c < NCHUNK; ++c) {
        const size_t idx = (size_t)c * NN + row;
        const float v = pposv[idx]; const int i = pposi[idx];
        if (v > bpv || (v == bpv && i < bpi)) { bpv = v; bpi = i; }
        const float u = pnegv[idx]; const int j = pnegi[idx];
        if (u < bnv || (u == bnv && j < bni)) { bnv = u; bni = j; }
    }
    hp[row] = bpi;
    hn[row] = bni;
}

// ---------------------------------------------------------------------------
// Kernel C: per-row triplet terms in exact fp32. One wave per anchor row.
// ---------------------------------------------------------------------------
__global__ __launch_bounds__(256) void triplet_kernel(
    const float* __restrict__ embed,
    const int* __restrict__ hp, const int* __restrict__ hn,
    float* __restrict__ lossArr)
{
    const int wave = (blockIdx.x * blockDim.x + threadIdx.x) >> 5;
    const int lane = threadIdx.x & 31;
    if (wave >= NN) return;
    const float4 av = *(const float4*)(embed + (size_t)wave * DD + lane * 4);
    const int pidx = hp[wave];
    const int nidx = hn[wave];
    const float4 pw = *(const float4*)(embed + (size_t)pidx * DD + lane * 4);
    const float4 nw = *(const float4*)(embed + (size_t)nidx * DD + lane * 4);
    float sap = 0.0f, san = 0.0f, d;
    d = av.x - pw.x + EPSV; sap += d * d;
    d = av.y - pw.y + EPSV; sap += d * d;
    d = av.z - pw.z + EPSV; sap += d * d;
    d = av.w - pw.w + EPSV; sap += d * d;
    d = av.x - nw.x + EPSV; san += d * d;
    d = av.y - nw.y + EPSV; san += d * d;
    d = av.z - nw.z + EPSV; san += d * d;
    d = av.w - nw.w + EPSV; san += d * d;
#pragma unroll
    for (int off = 16; off > 0; off >>= 1) {
        sap += __shfl_xor(sap, off, 32);
        san += __shfl_xor(san, off, 32);
    }
    if (lane == 0) {
        const float dap = sqrtf(sap);
        const float dan = sqrtf(san);
        lossArr[wave] = fmaxf(dap - dan + MARGINV, 0.0f);
    }
}

// ---------------------------------------------------------------------------
// Kernel D: deterministic fixed-order mean reduction (single block).
// ---------------------------------------------------------------------------
__global__ __launch_bounds__(256) void mean_kernel(
    const float* __restrict__ lossArr, float* __restrict__ out)
{
    __shared__ float sm[256];
    const int t = threadIdx.x;
    float s = 0.0f;
    for (int i = t; i < NN; i += 256) s += lossArr[i];
    sm[t] = s;
    __syncthreads();
    for (int off = 128; off > 0; off >>= 1) {
        if (t < off) sm[t] += sm[t + off];
        __syncthreads();
    }
    if (t == 0) out[0] = sm[0] / (float)NN;
}

// ---------------------------------------------------------------------------
extern "C" void kernel_launch(void* const* d_in, const int* in_sizes, int n_in,
                              void* d_out, int out_size, void* d_ws, size_t ws_size,
                              hipStream_t stream)
{
    const float* embed  = (const float*)d_in[0];
    const int*   labels = (const int*)d_in[1];

    char* ws = (char*)d_ws;
    float* normsq  = (float*)ws;                 ws += (size_t)NN * 4;
    int*   hp      = (int*)ws;                   ws += (size_t)NN * 4;
    int*   hn      = (int*)ws;                   ws += (size_t)NN * 4;
    float* lossArr = (float*)ws;                 ws += (size_t)NN * 4;
    float* pposv   = (float*)ws;                 ws += (size_t)NCHUNK * NN * 4;
    int*   pposi   = (int*)ws;                   ws += (size_t)NCHUNK * NN * 4;
    float* pnegv   = (float*)ws;                 ws += (size_t)NCHUNK * NN * 4;
    int*   pnegi   = (int*)ws;

    row_norms_kernel<<<NN / 8, 256, 0, stream>>>(embed, normsq);
    mine_kernel<<<dim3(NN / 128, NCHUNK), 256, 0, stream>>>(
        embed, labels, normsq, pposv, pposi, pnegv, pnegi);
    combine_kernel<<<NN / 256, 256, 0, stream>>>(pposv, pposi, pnegv, pnegi, hp, hn);
    triplet_kernel<<<NN / 8, 256, 0, stream>>>(embed, hp, hn, lossArr);
    mean_kernel<<<1, 256, 0, stream>>>(lossArr, (float*)d_out);
}